// GCNLayer_15779709845547
// MI455X (gfx1250) — compile-verified
//
#include <hip/hip_runtime.h>
#include <hip/hip_bf16.h>

typedef __attribute__((ext_vector_type(2))) float v2f;
typedef __attribute__((ext_vector_type(8))) float v8f;

#define IN_DIM 128
#define OUT_DIM 128
#define XS_LD 132   // 16x128 f32 tile with padded pitch: 132 % 64 banks = 4 -> conflict-free

__device__ __forceinline__ void atomAddF(float* p, float v) {
    // hardware global_atomic_add_f32 (non-returning), no CAS loop
    unsafeAtomicAdd(p, v);
}

// ---------------- zero workspace ----------------
__global__ void zero_kernel(float4* __restrict__ p, long n4) {
    long i = (long)blockIdx.x * blockDim.x + threadIdx.x;
    long stride = (long)gridDim.x * blockDim.x;
    float4 z = {0.f, 0.f, 0.f, 0.f};
    for (; i < n4; i += stride) p[i] = z;
}

// ---------------- in-degree ----------------
__global__ void deg_kernel(const int* __restrict__ dst, float* __restrict__ deg, int n_edges) {
    int i = blockIdx.x * blockDim.x + threadIdx.x;
    if (i < n_edges) atomAddF(&deg[dst[i]], 1.0f);
}

// ---------------- norm = clip(deg,1)^-1/2 (in place) ----------------
__global__ void norm_kernel(float* __restrict__ deg, int n_nodes) {
    int i = blockIdx.x * blockDim.x + threadIdx.x;
    if (i < n_nodes) deg[i] = rsqrtf(fmaxf(deg[i], 1.0f));
}

// ---------------- SpMM scatter: agg[dst] += h[src]*norm[src]; one wave32 per edge ----------------
__global__ void scatter_kernel(const float* __restrict__ h, const float* __restrict__ nrm,
                               const int* __restrict__ src, const int* __restrict__ dst,
                               float* __restrict__ agg, int n_edges) {
    int lane = threadIdx.x & 31;
    int e = blockIdx.x * (blockDim.x >> 5) + (threadIdx.x >> 5);
    if (e >= n_edges) return;
    int s = src[e];
    int d = dst[e];
    float nv = nrm[s];
    const float4* hp = (const float4*)(h + (size_t)s * IN_DIM);
    float4 v = hp[lane];                       // lane covers 4 consecutive floats, 32*4 = 128
    float* ap = agg + (size_t)d * IN_DIM + lane * 4;
    atomAddF(ap + 0, v.x * nv);
    atomAddF(ap + 1, v.y * nv);
    atomAddF(ap + 2, v.z * nv);
    atomAddF(ap + 3, v.w * nv);
}

// ---------------- GEMM: out = (agg*norm) @ W^T + b, via V_WMMA_F32_16X16X4_F32 ----------------
// Block: 256 threads = 8 waves. Block computes rows [row0, row0+16); wave w computes cols [16w, 16w+16).
__global__ void gemm_kernel(const float* __restrict__ agg, const float* __restrict__ nrm,
                            const float* __restrict__ W, const float* __restrict__ bias,
                            float* __restrict__ out, int n_nodes) {
    __shared__ float xs[16 * XS_LD];
    const int tid = threadIdx.x;
    const int row0 = blockIdx.x * 16;

    // Stage X tile = agg[row0:row0+16, :] * norm into LDS. 256 threads x 8 floats.
    {
        int r = tid >> 4;            // 0..15
        int c = (tid & 15) * 8;      // 0..120
        int row = row0 + r;
        float4 v0 = {0.f, 0.f, 0.f, 0.f}, v1 = v0;
        float nv = 0.f;
        if (row < n_nodes) {
            nv = nrm[row];
            const float4* sp = (const float4*)(agg + (size_t)row * IN_DIM + c);
            v0 = sp[0];
            v1 = sp[1];
        }
        float* xp = xs + r * XS_LD + c;
        xp[0] = v0.x * nv; xp[1] = v0.y * nv; xp[2] = v0.z * nv; xp[3] = v0.w * nv;
        xp[4] = v1.x * nv; xp[5] = v1.y * nv; xp[6] = v1.z * nv; xp[7] = v1.w * nv;
    }
    __syncthreads();

    const int lane = tid & 31;
    const int wave = tid >> 5;
    const int n0 = wave * 16;        // output column tile
    const int half = lane >> 4;      // 0 or 1
    const int lidx = lane & 15;      // 0..15

    v8f acc = {};
    // A frag: lanes 0-15 hold (M=lidx, K=k0..k0+1); lanes 16-31 hold (M=lidx, K=k0+2..k0+3).
    // B frag mirrors by N: B[k][n] = W[n][k] (row-major W [out,in]).
    const float* xrow = xs + lidx * XS_LD;
    const float* wrow = W + (size_t)(n0 + lidx) * IN_DIM;
#pragma unroll
    for (int k0 = 0; k0 < IN_DIM; k0 += 4) {
        int ka = k0 + 2 * half;
        v2f a; a.x = xrow[ka];  a.y = xrow[ka + 1];
        v2f bb; bb.x = wrow[ka]; bb.y = wrow[ka + 1];
        acc = __builtin_amdgcn_wmma_f32_16x16x4_f32(
            /*neg_a=*/false, a, /*neg_b=*/false, bb,
            /*c_mod=*/(short)0, acc, /*reuse_a=*/false, /*reuse_b=*/false);
    }

    float bv = bias[n0 + lidx];
#pragma unroll
    for (int j = 0; j < 8; ++j) {
        int row = row0 + j + 8 * half;   // D layout: VGPR j, M = j + 8*(lane/16), N = lane%16
        if (row < n_nodes) out[(size_t)row * OUT_DIM + n0 + lidx] = acc[j] + bv;
    }
}

extern "C" void kernel_launch(void* const* d_in, const int* in_sizes, int n_in,
                              void* d_out, int out_size, void* d_ws, size_t ws_size,
                              hipStream_t stream) {
    const float* h   = (const float*)d_in[0];
    const float* W   = (const float*)d_in[1];
    const float* b   = (const float*)d_in[2];
    const int* esrc  = (const int*)d_in[3];
    const int* edst  = (const int*)d_in[4];
    float* out = (float*)d_out;

    const int n_nodes = in_sizes[0] / IN_DIM;
    const int n_edges = in_sizes[3];

    // workspace layout: [norm: n_nodes floats][agg: n_nodes*IN_DIM floats]
    float* ws = (float*)d_ws;
    size_t agg_off = ((size_t)n_nodes + 3) & ~(size_t)3;   // keep 16B alignment
    float* nrm = ws;
    float* agg = ws + agg_off;

    // 1) zero norm + agg
    long n4 = (long)(agg_off + (size_t)n_nodes * IN_DIM) / 4;
    int zblocks = (int)((n4 + 255) / 256);
    if (zblocks > 8192) zblocks = 8192;
    zero_kernel<<<zblocks, 256, 0, stream>>>((float4*)ws, n4);

    // 2) degree
    deg_kernel<<<(n_edges + 255) / 256, 256, 0, stream>>>(edst, nrm, n_edges);

    // 3) norm (in place over deg)
    norm_kernel<<<(n_nodes + 255) / 256, 256, 0, stream>>>(nrm, n_nodes);

    // 4) edge scatter: one wave32 per edge, 8 edges per 256-thread block
    scatter_kernel<<<(n_edges + 7) / 8, 256, 0, stream>>>(h, nrm, esrc, edst, agg, n_edges);

    // 5) WMMA GEMM + bias
    gemm_kernel<<<(n_nodes + 15) / 16, 256, 0, stream>>>(agg, nrm, W, b, out, n_nodes);
}